// SinusoidalEncoder_20633022890154
// MI455X (gfx1250) — compile-verified
//
#include <hip/hip_runtime.h>
#include <hip/hip_bf16.h>

// Sinusoidal positional encoder, MI455X (gfx1250).
//
// out[pos, 2i]   = sin(x[pos,0] * 2^(i*10/128))
// out[pos, 2i+1] = cos(x[pos,1] * 2^(i*10/128)),  i = 0..127, pos = 0..262143
//
// Store-bandwidth bound: 256 MB written once -> ~11.5 us floor at 23.3 TB/s.
// Strategy: one thread owns one 16-byte output chunk (2 sin/cos pairs),
// frequencies hoisted out of the grid-stride loop (quad index is invariant
// because the stride is a multiple of 64), non-temporal global_store_b128
// for the streaming output so the 256 MB doesn't rinse through the caches.

typedef float v4f __attribute__((ext_vector_type(4)));

__global__ __launch_bounds__(256) void sinenc_kernel(const float* __restrict__ x,
                                                     float* __restrict__ out,
                                                     int nPos) {
    const int tid    = blockIdx.x * 256 + threadIdx.x;
    const int stride = gridDim.x * 256;            // multiple of 64 -> quad invariant
    const int quad   = tid & 63;                   // which float4 of the 64 per position

    // freqs for pair indices (2*quad, 2*quad+1): f = exp2(pair * 10/128)
    const float LOG2_STEP = 10.0f / 128.0f;        // 0.078125
    const float f0 = __builtin_amdgcn_exp2f((float)(2 * quad)     * LOG2_STEP); // v_exp_f32
    const float f1 = __builtin_amdgcn_exp2f((float)(2 * quad + 1) * LOG2_STEP); // v_exp_f32

    const int totalQuads = nPos << 6;              // 64 float4 per position
    v4f* __restrict__ out4 = (v4f*)out;

    for (int g = tid; g < totalQuads; g += stride) {
        const int pos = g >> 6;
        const float x0 = x[pos * 3 + 0];           // tiny input, hot in WGP$/L2
        const float x1 = x[pos * 3 + 1];

        v4f v;
        v.x = __sinf(x0 * f0);                     // v_sin_f32
        v.y = __cosf(x1 * f0);                     // v_cos_f32
        v.z = __sinf(x0 * f1);
        v.w = __cosf(x1 * f1);

        // streaming write-once output: TH=NT global_store_b128
        __builtin_nontemporal_store(v, out4 + g);
    }
}

extern "C" void kernel_launch(void* const* d_in, const int* in_sizes, int n_in,
                              void* d_out, int out_size, void* d_ws, size_t ws_size,
                              hipStream_t stream) {
    const float* x = (const float*)d_in[0];
    float* out = (float*)d_out;

    const int nPos = in_sizes[0] / 3;              // 32*8192 = 262144
    const int totalQuads = nPos << 6;              // 16,777,216 float4 stores

    // ~8 chunks per thread; plenty of waves for latency hiding.
    int blocks = (totalQuads + 256 * 8 - 1) / (256 * 8);
    if (blocks < 1) blocks = 1;

    sinenc_kernel<<<blocks, 256, 0, stream>>>(x, out, nPos);
}